// BatchDynamicSoftLabelAssigner_2740189135194
// MI455X (gfx1250) — compile-verified
//
#include <hip/hip_runtime.h>
#include <hip/hip_bf16.h>
#include <math.h>
#include <float.h>

// BatchDynamicSoftLabelAssigner for MI455X (gfx1250, wave32).
// B=32, N=8400, M=60, C=80. Cost tensor (61.5MB) lives in workspace and is
// L2-resident (192MB L2). WMMA f32 16x16x4 computes the prior-center x
// gt-center dot products (the GEMM-shaped fragment of the soft-center prior).

#define TOPK     13
#define INF_COST 1.0e8f
#define IOU_EPSF 1e-6f
#define LOG_EPSF 1e-7f
#define IOU_W    3.0f
#define LOG2_10  3.3219280948873623f

typedef __attribute__((ext_vector_type(2))) float v2f;
typedef __attribute__((ext_vector_type(8))) float v8f;

__device__ __forceinline__ float iou_pair(float px1, float py1, float px2, float py2, float pa,
                                          float gx1, float gy1, float gx2, float gy2, float ga) {
  float ix1 = fmaxf(px1, gx1), iy1 = fmaxf(py1, gy1);
  float ix2 = fminf(px2, gx2), iy2 = fminf(py2, gy2);
  float iw = fmaxf(ix2 - ix1, 0.f), ih = fmaxf(iy2 - iy1, 0.f);
  float inter = iw * ih;
  float uni = fmaxf(pa + ga - inter, IOU_EPSF);
  return inter / uni;
}

// ---------------------------------------------------------------- pass 0
// valid_mask[b,n] = OR_m (prior center strictly inside gt box m AND pad flag);
// also zero the per-(b,n) 64-bit matching masks.
__global__ void k_valid_init(const float* __restrict__ priors,
                             const float* __restrict__ gt_bboxes,
                             const float* __restrict__ pad,
                             float* __restrict__ valid,
                             unsigned long long* __restrict__ mmask,
                             int B, int N, int M) {
  int idx = blockIdx.x * blockDim.x + threadIdx.x;
  if (idx >= B * N) return;
  int b = idx / N, n = idx - b * N;
  float px = priors[n * 4 + 0], py = priors[n * 4 + 1];
  int any = 0;
  for (int m = 0; m < M; ++m) {
    const float* g = gt_bboxes + ((size_t)b * M + m) * 4;
    float mn = fminf(fminf(px - g[0], py - g[1]), fminf(g[2] - px, g[3] - py));
    any |= (mn > 0.f) && (pad[b * M + m] > 0.f);
  }
  valid[idx] = any ? 1.f : 0.f;
  mmask[idx] = 0ull;
}

// ---------------------------------------------------------------- pass 1
// One wave per (batch, 16-prior tile). Four V_WMMA_F32_16X16X4_F32 cover the
// (padded to 64) gt dimension: A = gt centers (16x4, K=[gx,gy,0,0]),
// B = prior centers (4x16, col n = [px,py,0,0]) -> D[m][n] = g.p.
// dist^2 = |p|^2 + |g|^2 - 2 D. Then full cost per element, stored [b][m][n].
__global__ void __launch_bounds__(32)
k_cost_wmma(const float* __restrict__ pred_bboxes,
            const float* __restrict__ pred_scores,
            const float* __restrict__ priors,
            const int*   __restrict__ gt_labels,
            const float* __restrict__ gt_bboxes,
            const float* __restrict__ valid,
            float* __restrict__ cost,
            int B, int N, int M, int C) {
  __shared__ float sgx[64], sgy[64], sgn2[64], sga[64];
  __shared__ float sgb[64 * 4];
  __shared__ int   slab[64];
  int lane = threadIdx.x;
  int b = blockIdx.y;
  int n0 = blockIdx.x * 16;

  for (int m = lane; m < 64; m += 32) {      // requires M <= 64 (M=60)
    float x1 = 0, y1 = 0, x2 = 0, y2 = 0; int lb = 0;
    if (m < M) {
      const float* g = gt_bboxes + ((size_t)b * M + m) * 4;
      x1 = g[0]; y1 = g[1]; x2 = g[2]; y2 = g[3];
      lb = gt_labels[b * M + m];
    }
    float cx = (x1 + x2) * 0.5f, cy = (y1 + y2) * 0.5f;
    sgx[m] = cx; sgy[m] = cy; sgn2[m] = cx * cx + cy * cy;
    sga[m] = (x2 - x1) * (y2 - y1);
    sgb[m * 4 + 0] = x1; sgb[m * 4 + 1] = y1; sgb[m * 4 + 2] = x2; sgb[m * 4 + 3] = y2;
    slab[m] = lb;
  }
  __syncthreads();

  int nl = n0 + (lane & 15);
  int n  = (nl < N) ? nl : (N - 1);          // N=8400 is a multiple of 16; clamp for safety
  float px = priors[n * 4 + 0], py = priors[n * 4 + 1], stride = priors[n * 4 + 2];
  float pn2 = px * px + py * py;
  const float* pb = pred_bboxes + ((size_t)b * N + n) * 4;
  float px1 = pb[0], py1 = pb[1], px2 = pb[2], py2v = pb[3];
  float pa = (px2 - px1) * (py2v - py1);
  float vm = valid[(size_t)b * N + n];
  const float* ps = pred_scores + ((size_t)b * N + n) * (size_t)C;

#pragma unroll
  for (int mt = 0; mt < 4; ++mt) {
    int m0 = mt * 16;
    int ml = m0 + (lane & 15);
    v2f av, bv;
    av.x = (lane < 16) ? sgx[ml] : 0.f;      // A: K=0/1 rows in lanes 0-15, K=2/3 zero
    av.y = (lane < 16) ? sgy[ml] : 0.f;
    bv.x = (lane < 16) ? px : py;            // B: row K=0 (px) lanes 0-15, K=1 (py) lanes 16-31
    bv.y = 0.f;
    v8f cacc = {0.f, 0.f, 0.f, 0.f, 0.f, 0.f, 0.f, 0.f};
    v8f dmat = __builtin_amdgcn_wmma_f32_16x16x4_f32(
        false, av, false, bv, (short)0, cacc, false, false);

#pragma unroll
    for (int r = 0; r < 8; ++r) {
      int m = m0 + r + ((lane >> 4) << 3);   // D layout: lanes 16-31 hold rows M=r+8
      if (m < M && nl < N) {
        float dot  = dmat[r];
        float d2   = fmaxf(pn2 + sgn2[m] - 2.f * dot, 0.f);
        float dist = sqrtf(d2) / stride;
        float soft = exp2f((dist - 3.0f) * LOG2_10);       // 10^(dist-3)
        float iou  = iou_pair(px1, py1, px2, py2v, pa,
                              sgb[m * 4 + 0], sgb[m * 4 + 1],
                              sgb[m * 4 + 2], sgb[m * 4 + 3], sga[m]);
        float iouc = -logf(iou + LOG_EPSF) * IOU_W;
        float t    = ps[slab[m]];                          // logit gather (dominant traffic)
        float sig  = 1.f / (1.f + expf(-t));
        float sc   = iou - sig;
        float bce  = fmaxf(t, 0.f) - t * iou + log1pf(expf(-fabsf(t)));
        float cval = (vm > 0.f) ? (bce * sc * sc + iouc + soft) : INF_COST;
        cost[((size_t)b * M + m) * N + n] = cval;
      }
    }
  }
}

// ---------------------------------------------------------------- pass 2
// One block per (b,m). Phase 1: top-13 IoUs -> dynamic_k. Phase 2: top-13
// smallest (cost,index) with stable tie-break; thread 0 sets mask bits.
__global__ void __launch_bounds__(256)
k_select(const float* __restrict__ pred_bboxes,
         const float* __restrict__ gt_bboxes,
         const float* __restrict__ pad,
         const float* __restrict__ cost,
         unsigned long long* __restrict__ mmask,
         int B, int N, int M) {
  __shared__ float sV[256 * TOPK];
  __shared__ int   sIx[256 * TOPK];
  int m = blockIdx.x, b = blockIdx.y;
  if (pad[b * M + m] <= 0.f) return;         // uniform per block
  int t = threadIdx.x;

  const float* g = gt_bboxes + ((size_t)b * M + m) * 4;
  float gx1 = g[0], gy1 = g[1], gx2 = g[2], gy2 = g[3];
  float ga = (gx2 - gx1) * (gy2 - gy1);

  // ---- phase 1: per-thread top-13 IoUs (descending), register resident ----
  float top[TOPK];
#pragma unroll
  for (int j = 0; j < TOPK; ++j) top[j] = -1.f;
  for (int n = t; n < N; n += 256) {
    const float* pb = pred_bboxes + ((size_t)b * N + n) * 4;
    float pa = (pb[2] - pb[0]) * (pb[3] - pb[1]);
    float cur = iou_pair(pb[0], pb[1], pb[2], pb[3], pa, gx1, gy1, gx2, gy2, ga);
#pragma unroll
    for (int j = 0; j < TOPK; ++j) {         // conditional-swap insertion
      float o = top[j];
      bool gtc = cur > o;
      top[j] = gtc ? cur : o;
      cur    = gtc ? o : cur;
    }
  }
#pragma unroll
  for (int j = 0; j < TOPK; ++j) sV[t * TOPK + j] = top[j];
  for (int s = 128; s >= 1; s >>= 1) {       // tournament merge of sorted 13-lists
    __syncthreads();
    if (t < s) {
      float outv[TOPK];
      int i = 0, j = 0;
#pragma unroll
      for (int q = 0; q < TOPK; ++q) {
        float a = sV[t * TOPK + i], c = sV[(t + s) * TOPK + j];
        bool ta = a >= c;
        outv[q] = ta ? a : c;
        i += ta; j += !ta;
      }
#pragma unroll
      for (int q = 0; q < TOPK; ++q) sV[t * TOPK + q] = outv[q];
    }
  }
  __syncthreads();
  int dk = 1;
  if (t == 0) {
    float s13 = 0.f;
#pragma unroll
    for (int j = 0; j < TOPK; ++j) s13 += sV[j];
    dk = (int)s13;                           // truncation matches astype(int32)
    if (dk < 1) dk = 1;
    if (dk > TOPK) dk = TOPK;
  }
  __syncthreads();

  // ---- phase 2: top-13 smallest (cost, index), stable tie-break ----
  float tc[TOPK]; int ti[TOPK];
#pragma unroll
  for (int j = 0; j < TOPK; ++j) { tc[j] = FLT_MAX; ti[j] = 0x7fffffff; }
  const float* crow = cost + ((size_t)b * M + m) * N;
  for (int n = t; n < N; n += 256) {
    float cv = crow[n]; int ci = n;
#pragma unroll
    for (int j = 0; j < TOPK; ++j) {
      float oc = tc[j]; int oi = ti[j];
      bool lt = (cv < oc) || ((cv == oc) && (ci < oi));
      tc[j] = lt ? cv : oc; ti[j] = lt ? ci : oi;
      cv    = lt ? oc : cv; ci    = lt ? oi : ci;
    }
  }
#pragma unroll
  for (int j = 0; j < TOPK; ++j) { sV[t * TOPK + j] = tc[j]; sIx[t * TOPK + j] = ti[j]; }
  for (int s = 128; s >= 1; s >>= 1) {
    __syncthreads();
    if (t < s) {
      float oc[TOPK]; int oi[TOPK];
      int i = 0, j = 0;
#pragma unroll
      for (int q = 0; q < TOPK; ++q) {
        float ca = sV[t * TOPK + i];       int ia = sIx[t * TOPK + i];
        float cb = sV[(t + s) * TOPK + j]; int ib = sIx[(t + s) * TOPK + j];
        bool ta = (ca < cb) || ((ca == cb) && (ia < ib));
        oc[q] = ta ? ca : cb; oi[q] = ta ? ia : ib;
        i += ta; j += !ta;
      }
#pragma unroll
      for (int q = 0; q < TOPK; ++q) { sV[t * TOPK + q] = oc[q]; sIx[t * TOPK + q] = oi[q]; }
    }
  }
  __syncthreads();
  if (t == 0) {
    for (int j = 0; j < dk; ++j) {
      int n = sIx[j];
      if (n >= 0 && n < N)
        atomicOr(&mmask[(size_t)b * N + n], 1ull << m);
    }
  }
}

// ---------------------------------------------------------------- pass 3
// Per (b,n): resolve multi-match via argmin over m, gather outputs.
__global__ void k_finalize(const float* __restrict__ pred_bboxes,
                           const int* __restrict__ gt_labels,
                           const float* __restrict__ gt_bboxes,
                           const float* __restrict__ cost,
                           const unsigned long long* __restrict__ mmask,
                           float* __restrict__ out,
                           int B, int N, int M, int C) {
  int idx = blockIdx.x * blockDim.x + threadIdx.x;
  int BN = B * N;
  if (idx >= BN) return;
  int b = idx / N, n = idx - b * N;
  unsigned long long mk = mmask[idx];
  if (__popcll(mk) > 1) {
    float best = FLT_MAX; int bm = 0;
    for (int m = 0; m < M; ++m) {           // coalesced over n per m step
      float c = cost[((size_t)b * M + m) * N + n];
      if (c < best) { best = c; bm = m; }   // first-min, like jnp.argmin
    }
    mk = 1ull << bm;
  }
  bool fg = (mk != 0ull);
  int mg = fg ? (__ffsll((unsigned long long)mk) - 1) : 0;
  float lab = fg ? (float)gt_labels[b * M + mg] : (float)C;
  float bx0 = 0.f, bx1 = 0.f, bx2 = 0.f, bx3 = 0.f, metric = 0.f;
  if (fg) {
    const float* g = gt_bboxes + ((size_t)b * M + mg) * 4;
    bx0 = g[0]; bx1 = g[1]; bx2 = g[2]; bx3 = g[3];
    const float* pb = pred_bboxes + ((size_t)b * N + n) * 4;
    float pa = (pb[2] - pb[0]) * (pb[3] - pb[1]);
    float ga = (bx2 - bx0) * (bx3 - bx1);
    metric = iou_pair(pb[0], pb[1], pb[2], pb[3], pa, bx0, bx1, bx2, bx3, ga);
  }
  // outputs (f32), tuple order: labels, weights, bboxes, metrics
  out[idx]                 = lab;
  out[BN + idx]            = 1.0f;
  out[2 * BN + idx * 4 + 0] = bx0;
  out[2 * BN + idx * 4 + 1] = bx1;
  out[2 * BN + idx * 4 + 2] = bx2;
  out[2 * BN + idx * 4 + 3] = bx3;
  out[6 * BN + idx]        = metric;
}

extern "C" void kernel_launch(void* const* d_in, const int* in_sizes, int n_in,
                              void* d_out, int out_size, void* d_ws, size_t ws_size,
                              hipStream_t stream) {
  const float* pred_bboxes = (const float*)d_in[0];
  const float* pred_scores = (const float*)d_in[1];
  const float* priors      = (const float*)d_in[2];
  const int*   gt_labels   = (const int*)d_in[3];
  const float* gt_bboxes   = (const float*)d_in[4];
  const float* pad         = (const float*)d_in[5];

  int N = in_sizes[2] / 4;                   // 8400
  int B = in_sizes[0] / (N * 4);             // 32
  int M = in_sizes[4] / (B * 4);             // 60 (must be <= 64)
  int C = in_sizes[1] / (B * N);             // 80

  // workspace layout: cost[B*M*N] f32 | valid[B*N] f32 | mmask[B*N] u64 (8B aligned)
  float* cost  = (float*)d_ws;
  float* valid = cost + (size_t)B * M * N;
  size_t off   = (size_t)B * M * N + (size_t)B * N;
  off = (off + 1) & ~(size_t)1;              // round to 8-byte boundary
  unsigned long long* mmask = (unsigned long long*)((float*)d_ws + off);

  int BN = B * N;
  k_valid_init<<<(BN + 255) / 256, 256, 0, stream>>>(
      priors, gt_bboxes, pad, valid, mmask, B, N, M);

  dim3 g1((N + 15) / 16, B);
  k_cost_wmma<<<g1, 32, 0, stream>>>(
      pred_bboxes, pred_scores, priors, gt_labels, gt_bboxes, valid, cost, B, N, M, C);

  dim3 g2(M, B);
  k_select<<<g2, 256, 0, stream>>>(
      pred_bboxes, gt_bboxes, pad, cost, mmask, B, N, M);

  k_finalize<<<(BN + 255) / 256, 256, 0, stream>>>(
      pred_bboxes, gt_labels, gt_bboxes, cost, mmask, (float*)d_out, B, N, M, C);
}